// GraphNet_3959959846913
// MI455X (gfx1250) — compile-verified
//
#include <hip/hip_runtime.h>
#include <hip/hip_bf16.h>

typedef __attribute__((ext_vector_type(2))) float v2f;
typedef __attribute__((ext_vector_type(8))) float v8f;

#define N_NODES_C 100000
#define N_EDGES_C 1600000
#define N_GRAPHS_C 64
#define IN_DIM_C 128
#define HID_DIM_C 64
#define OUT_DIM_C 32

static inline int cdiv(long long a, long long b) { return (int)((a + b - 1) / b); }

// ---------------------------------------------------------------------------
// 1) Degree: deg[dst] += 1 per edge (self-loop folded in analytically).
// ---------------------------------------------------------------------------
__global__ void degree_kernel(const int* __restrict__ dst, float* __restrict__ deg,
                              int nedges) {
    int e = blockIdx.x * blockDim.x + threadIdx.x;
    if (e < nedges) atomicAdd(&deg[dst[e]], 1.0f);
}

__global__ void dinv_kernel(float* __restrict__ deg, int n) {
    int i = blockIdx.x * blockDim.x + threadIdx.x;
    if (i < n) deg[i] = __frsqrt_rn(deg[i] + 1.0f);  // deg+1 >= 1
}

// ---------------------------------------------------------------------------
// 2) GEMM via V_WMMA_F32_16X16X4_F32.  One wave produces a 16-row x N tile.
//    W is staged into LDS pair-interleaved: entry (p, n) holds the float2
//    (W[2p][n], W[2p+1][n]).  The WMMA B operand (two K-adjacent values for
//    one column) is then a single aligned ds_load_b64 into an even VGPR pair,
//    avoiding register-marshalling moves in the inner loop.
//    REQUIREMENT: nrows % 16 == 0 (holds: 100000 = 6250*16) -> no tail guards,
//    EXEC stays all-ones for every wave that enters the WMMA loop.
//    Layouts per ISA tables:
//      A 16x4 : lanes 0-15 M=lane,  v[0]=K0 v[1]=K1 ; lanes 16-31 K2/K3
//      B 4x16 : row-striped across lanes (VGPR0: K0 | K2, VGPR1: K1 | K3)
//      D 16x16: VGPR i -> M=i (lanes 0-15) / M=i+8 (lanes 16-31), N=lane&15
// ---------------------------------------------------------------------------
template <int K, int N>
__global__ void gemm_scale_kernel(const float* __restrict__ X,
                                  const float* __restrict__ W,
                                  const float* __restrict__ dinv,
                                  float* __restrict__ out, int nrows) {
    __shared__ v2f sW[(K / 2) * N];  // pair-interleaved W
    for (int idx = threadIdx.x; idx < (K / 2) * N; idx += blockDim.x) {
        const int p = idx / N;
        const int n = idx - p * N;
        v2f w;
        w.x = W[(2 * p) * N + n];
        w.y = W[(2 * p + 1) * N + n];
        sW[idx] = w;
    }
    __syncthreads();

    const int lane = threadIdx.x & 31;
    const int wave = (blockIdx.x * blockDim.x + threadIdx.x) >> 5;
    const int row0 = wave * 16;
    if (row0 >= nrows) return;  // wave-uniform: EXEC all-ones below

    const int r  = lane & 15;   // M (A/D rows) or N column (B/D)
    const int kh = lane >> 4;   // K half-select (0 or 1)
    const float* xrow = X + (size_t)(row0 + r) * K + 2 * kh;

    constexpr int NT = N / 16;
    v8f acc[NT] = {};

#pragma unroll 4
    for (int k0 = 0; k0 < K; k0 += 4) {
        v2f a = *reinterpret_cast<const v2f*>(xrow + k0);  // b64 load
        const int p = (k0 >> 1) + kh;                      // K-pair index
#pragma unroll
        for (int t = 0; t < NT; ++t) {
            v2f b = sW[p * N + t * 16 + r];                // ds_load_b64
            acc[t] = __builtin_amdgcn_wmma_f32_16x16x4_f32(
                false, a, false, b, (short)0, acc[t], false, false);
        }
    }

    // Preload the 8 dinv values this lane scales with (consecutive rows).
    const int rbase = row0 + 8 * kh;
    float dv[8];
#pragma unroll
    for (int i = 0; i < 8; ++i) dv[i] = dinv[rbase + i];

#pragma unroll
    for (int t = 0; t < NT; ++t) {
#pragma unroll
        for (int i = 0; i < 8; ++i) {
            out[(size_t)(rbase + i) * N + t * 16 + r] = acc[t][i] * dv[i];
        }
    }
}

// ---------------------------------------------------------------------------
// 3) Edge scatter: agg[dst][c..c+3] += hs[src][c..c+3].
//    LPE lanes per edge; each lane moves a float4 (coalesced 128-bit gather,
//    non-returning global_atomic_add_f32 resolves in L2).
// ---------------------------------------------------------------------------
template <int CH, int LPE>
__global__ void scatter_add_kernel(const int* __restrict__ src,
                                   const int* __restrict__ dst,
                                   const float* __restrict__ hs,
                                   float* __restrict__ agg, int nedges) {
    int t = blockIdx.x * blockDim.x + threadIdx.x;
    int e = t / LPE;
    if (e >= nedges) return;
    int c = (t % LPE) * 4;
    int s = src[e], d = dst[e];
    const float4 v = *reinterpret_cast<const float4*>(hs + (long long)s * CH + c);
    float* p = agg + (long long)d * CH + c;
    atomicAdd(p + 0, v.x);
    atomicAdd(p + 1, v.y);
    atomicAdd(p + 2, v.z);
    atomicAdd(p + 3, v.w);
}

// ---------------------------------------------------------------------------
// 4) Finalize layer 1: out = relu(agg * dinv[row] + bias)
// ---------------------------------------------------------------------------
template <int CH>
__global__ void finalize_relu_kernel(const float* __restrict__ agg,
                                     const float* __restrict__ dinv,
                                     const float* __restrict__ bias,
                                     float* __restrict__ out, int nnodes) {
    long long idx = (long long)blockIdx.x * blockDim.x + threadIdx.x;
    if (idx >= (long long)nnodes * CH) return;
    int i = (int)(idx / CH);
    int c = (int)(idx % CH);
    float v = agg[idx] * dinv[i] + bias[c];
    out[idx] = v > 0.0f ? v : 0.0f;
}

// ---------------------------------------------------------------------------
// 5) Finalize layer 2 fused with mean-pool accumulation.
// ---------------------------------------------------------------------------
__global__ void finalize_pool_kernel(const float* __restrict__ agg,
                                     const float* __restrict__ dinv,
                                     const float* __restrict__ bias,
                                     const int* __restrict__ batch,
                                     float* __restrict__ pool,
                                     float* __restrict__ cnt, int nnodes) {
    long long idx = (long long)blockIdx.x * blockDim.x + threadIdx.x;
    if (idx >= (long long)nnodes * OUT_DIM_C) return;
    int i = (int)(idx >> 5);  // OUT_DIM_C == 32
    int c = (int)(idx & 31);
    float v = agg[idx] * dinv[i] + bias[c];
    int g = batch[i];
    atomicAdd(&pool[g * OUT_DIM_C + c], v);
    if (c == 0) atomicAdd(&cnt[g], 1.0f);
}

__global__ void pool_div_kernel(const float* __restrict__ pool,
                                const float* __restrict__ cnt,
                                float* __restrict__ out) {
    int idx = blockIdx.x * blockDim.x + threadIdx.x;
    if (idx >= N_GRAPHS_C * OUT_DIM_C) return;
    int g = idx >> 5;
    out[idx] = pool[idx] / fmaxf(cnt[g], 1.0f);
}

// ---------------------------------------------------------------------------
extern "C" void kernel_launch(void* const* d_in, const int* in_sizes, int n_in,
                              void* d_out, int out_size, void* d_ws, size_t ws_size,
                              hipStream_t stream) {
    const float* x     = (const float*)d_in[0];
    const int*   ei    = (const int*)d_in[1];   // [2, E] flat: src row then dst row
    const int*   batch = (const int*)d_in[2];
    const float* W1    = (const float*)d_in[3];
    const float* b1    = (const float*)d_in[4];
    const float* W2    = (const float*)d_in[5];
    const float* b2    = (const float*)d_in[6];
    float* out = (float*)d_out;

    const int NN = N_NODES_C;
    const int NE = in_sizes[1] / 2;  // == N_EDGES_C
    const int* src = ei;
    const int* dst = ei + NE;

    // Workspace layout (floats):
    float* dinv = (float*)d_ws;                   // NN
    float* bufA = dinv + NN;                      // NN*64 (hs1, then h1_relu)
    float* bufB = bufA + (size_t)NN * HID_DIM_C;  // NN*64 (agg1, then agg2)
    float* bufC = bufB + (size_t)NN * HID_DIM_C;  // NN*32 (hs2)
    float* pool = bufC + (size_t)NN * OUT_DIM_C;  // 64*32
    float* cnt  = pool + N_GRAPHS_C * OUT_DIM_C;  // 64

    const int B = 256;

    // --- degrees / normalization ---
    hipMemsetAsync(dinv, 0, sizeof(float) * NN, stream);
    hipMemsetAsync(pool, 0, sizeof(float) * (N_GRAPHS_C * OUT_DIM_C + N_GRAPHS_C),
                   stream);
    degree_kernel<<<cdiv(NE, B), B, 0, stream>>>(dst, dinv, NE);
    dinv_kernel<<<cdiv(NN, B), B, 0, stream>>>(dinv, NN);

    // --- layer 1: hs1 = dinv * (x @ W1) via WMMA (NN % 16 == 0) ---
    {
        int waves = NN / 16;
        gemm_scale_kernel<IN_DIM_C, HID_DIM_C>
            <<<cdiv((long long)waves * 32, B), B, 0, stream>>>(x, W1, dinv, bufA, NN);
    }
    // self-loop term: agg starts as hs1
    hipMemcpyAsync(bufB, bufA, sizeof(float) * (size_t)NN * HID_DIM_C,
                   hipMemcpyDeviceToDevice, stream);
    scatter_add_kernel<HID_DIM_C, HID_DIM_C / 4>
        <<<cdiv((long long)NE * (HID_DIM_C / 4), B), B, 0, stream>>>(src, dst, bufA,
                                                                     bufB, NE);
    finalize_relu_kernel<HID_DIM_C>
        <<<cdiv((long long)NN * HID_DIM_C, B), B, 0, stream>>>(bufB, dinv, b1, bufA,
                                                               NN);

    // --- layer 2: hs2 = dinv * (h1 @ W2) via WMMA ---
    {
        int waves = NN / 16;
        gemm_scale_kernel<HID_DIM_C, OUT_DIM_C>
            <<<cdiv((long long)waves * 32, B), B, 0, stream>>>(bufA, W2, dinv, bufC,
                                                               NN);
    }
    hipMemcpyAsync(bufB, bufC, sizeof(float) * (size_t)NN * OUT_DIM_C,
                   hipMemcpyDeviceToDevice, stream);
    scatter_add_kernel<OUT_DIM_C, OUT_DIM_C / 4>
        <<<cdiv((long long)NE * (OUT_DIM_C / 4), B), B, 0, stream>>>(src, dst, bufC,
                                                                     bufB, NE);

    // --- finalize + mean pool ---
    finalize_pool_kernel<<<cdiv((long long)NN * OUT_DIM_C, B), B, 0, stream>>>(
        bufB, dinv, b2, batch, pool, cnt, NN);
    pool_div_kernel<<<cdiv(N_GRAPHS_C * OUT_DIM_C, B), B, 0, stream>>>(pool, cnt, out);
}